// StateNeuron_8074538517094
// MI455X (gfx1250) — compile-verified
//
#include <hip/hip_runtime.h>
#include <hip/hip_bf16.h>
#include <cmath>

#define ORDER 256
#define IN_DIM 128
#define SEQ 8192
#define APAD 260   // padded LDS row stride (floats): 1040B, 16B aligned, avoids 32-way bank conflicts

typedef __attribute__((ext_vector_type(2))) float v2f;
typedef __attribute__((ext_vector_type(8))) float v8f;

// ---------------------------------------------------------------------------
// Kernel 1: BuT[t][r] = sum_k B[r][k] * u[k][t]   via V_WMMA_F32_16X16X4_F32
// One wave per 16x16 output tile. M = ORDER rows of B, N = SEQ columns of u.
// ---------------------------------------------------------------------------
__global__ __launch_bounds__(32) void bu_wmma_kernel(const float* __restrict__ Bm,
                                                     const float* __restrict__ u,
                                                     float* __restrict__ BuT) {
  const int n0   = blockIdx.x * 16;   // seq tile
  const int m0   = blockIdx.y * 16;   // order tile
  const int lane = threadIdx.x;       // 0..31 (wave32)
  const int half = lane >> 4;         // 0: K=k0..k0+1, 1: K=k0+2..k0+3
  const int l15  = lane & 15;

  v8f c = {0.f, 0.f, 0.f, 0.f, 0.f, 0.f, 0.f, 0.f};

  for (int k0 = 0; k0 < IN_DIM; k0 += 4) {
    // A-fragment (problem matrix B): 16x4 f32. lane l<16 -> M=l, VGPR0/1 = K=k0,k0+1
    //                                 lanes 16..31       -> M=l-16, K=k0+2,k0+3
    const v2f a = *(const v2f*)&Bm[(m0 + l15) * IN_DIM + k0 + 2 * half];  // 8B aligned

    // B-fragment (u): 4x16 f32. lane l<16 -> N=l, VGPR0/1 = K=k0,k0+1; lanes>=16 -> K=k0+2,k0+3
    v2f b;
    b.x = u[(k0 + 2 * half + 0) * SEQ + n0 + l15];
    b.y = u[(k0 + 2 * half + 1) * SEQ + n0 + l15];

    // 8-arg pattern: (neg_a, A, neg_b, B, c_mod, C, reuse_a, reuse_b)
    c = __builtin_amdgcn_wmma_f32_16x16x4_f32(false, a, false, b, (short)0, c,
                                              false, false);
  }

  // C/D layout: VGPR i -> row M = i (lanes 0-15) or i+8 (lanes 16-31), N = lane&15.
  // Store transposed so the scan kernel reads a contiguous 256-float row per step.
#pragma unroll
  for (int i = 0; i < 8; ++i) {
    const int m = i + 8 * half;
    BuT[(size_t)(n0 + l15) * ORDER + (m0 + m)] = c[i];
  }
}

// ---------------------------------------------------------------------------
// Kernel 2: Du[t] = sum_k D[k] * u[k][t]
// ---------------------------------------------------------------------------
__global__ __launch_bounds__(256) void du_kernel(const float* __restrict__ D,
                                                 const float* __restrict__ u,
                                                 float* __restrict__ Du) {
  const int t = blockIdx.x * blockDim.x + threadIdx.x;
  if (t < SEQ) {
    float s = 0.f;
#pragma unroll 8
    for (int k = 0; k < IN_DIM; ++k) s = fmaf(D[k], u[(size_t)k * SEQ + t], s);
    Du[t] = s;
  }
}

// ---------------------------------------------------------------------------
// Kernel 3: sequential scan. One workgroup, 256 threads (8 wave32s), A in LDS.
// LDS: A padded rows (256*260 floats) + h (256) + per-wave partials (8) ~= 267KB
// (CDNA5 WGP allows up to 320KB per workgroup).
// ---------------------------------------------------------------------------
__global__ __launch_bounds__(256) void scan_kernel(
    const float* __restrict__ A, const float* __restrict__ C,
    const float* __restrict__ L, const float* __restrict__ alpha,
    const float* __restrict__ y_obs, const float* __restrict__ noise,
    const float* __restrict__ delta_p, const float* __restrict__ BuT,
    const float* __restrict__ Du, float* __restrict__ out) {
  extern __shared__ float smem[];
  float* As  = smem;                       // ORDER * APAD
  float* hs  = smem + ORDER * APAD;        // ORDER (16B aligned: 256*260*4 % 16 == 0)
  float* red = hs + ORDER;                 // 8 per-wave partials

  const int r    = threadIdx.x;            // state row owned by this thread
  const int wid  = r >> 5;
  const int lane = r & 31;

  // Preload A: iteration i writes row i, columns = tid -> coalesced, conflict-free.
  for (int i = 0; i < ORDER; ++i) As[i * APAD + r] = A[(size_t)i * ORDER + r];

  const float Cr     = C[r];
  const float Lr     = L[r];
  const float ar     = alpha[r];
  const float delta  = delta_p[0];
  const float inv_pd = powf(delta, ar - 1.0f);  // 1 / delta^(1-alpha)

  hs[r] = 1.0f;  // h0 = ones
  __syncthreads();

  const float* Ar = &As[r * APAD];

  for (int t = 0; t < SEQ; ++t) {
    // ---- C.h : per-thread product, wave reduce, stash per-wave partial ----
    float cy = Cr * hs[r];
#pragma unroll
    for (int off = 16; off > 0; off >>= 1) cy += __shfl_xor(cy, off, 32);
    if (lane == 0) red[wid] = cy;

    // ---- dot = A[r,:] . h  (reads old h; float4 LDS traffic) ----
    float dot = 0.f;
#pragma unroll 8
    for (int k = 0; k < ORDER; k += 4) {
      const float4 hv = *(const float4*)&hs[k];  // same-address broadcast
      const float4 av = *(const float4*)&Ar[k];  // ds_load_b128, row stride 1040B
      dot = fmaf(av.x, hv.x, dot);
      dot = fmaf(av.y, hv.y, dot);
      dot = fmaf(av.z, hv.z, dot);
      dot = fmaf(av.w, hv.w, dot);
    }
    __syncthreads();  // partials visible; all reads of hs complete

    const float ylin = red[0] + red[1] + red[2] + red[3] +
                       red[4] + red[5] + red[6] + red[7];
    const float y   = ylin + Du[t] + noise[t];
    const float err = y_obs[t] - y;
    if (r == 0) out[t] = tanhf(y);

    // ---- nonlinear correction ----
    const float aerr = fabsf(err);
    float nl;
    if (aerr <= delta) nl = err * inv_pd;
    else               nl = copysignf(__expf(ar * __logf(aerr)), err);

    hs[r] = dot + BuT[(size_t)t * ORDER + r] + Lr * nl;
    __syncthreads();  // new h visible before next step
  }
}

// ---------------------------------------------------------------------------
extern "C" void kernel_launch(void* const* d_in, const int* in_sizes, int n_in,
                              void* d_out, int out_size, void* d_ws,
                              size_t ws_size, hipStream_t stream) {
  const float* u     = (const float*)d_in[0];  // [128, 8192]
  const float* y_obs = (const float*)d_in[1];  // [1, 8192]
  const float* noise = (const float*)d_in[2];  // [8192]
  const float* A     = (const float*)d_in[3];  // [256, 256]
  const float* B     = (const float*)d_in[4];  // [256, 128]
  const float* C     = (const float*)d_in[5];  // [1, 256]
  const float* D     = (const float*)d_in[6];  // [1, 128]
  const float* L     = (const float*)d_in[7];  // [256, 1]
  const float* alpha = (const float*)d_in[8];  // [256, 1]
  const float* delta = (const float*)d_in[9];  // scalar
  float* out = (float*)d_out;                  // [1, 8192]

  float* BuT = (float*)d_ws;                   // SEQ*ORDER floats (8 MB)
  float* Du  = BuT + (size_t)SEQ * ORDER;      // SEQ floats

  dim3 g1(SEQ / 16, ORDER / 16);               // 512 x 16 tiles, one wave each
  bu_wmma_kernel<<<g1, 32, 0, stream>>>(B, u, BuT);
  du_kernel<<<SEQ / 256, 256, 0, stream>>>(D, u, Du);

  const size_t smem = (size_t)(ORDER * APAD + ORDER + 8) * sizeof(float);
  scan_kernel<<<1, ORDER, smem, stream>>>(A, C, L, alpha, y_obs, noise, delta,
                                          BuT, Du, out);
}